// GATEncoder_81011673137266
// MI455X (gfx1250) — compile-verified
//
#include <hip/hip_runtime.h>
#include <hip/hip_bf16.h>
#include <stdint.h>

typedef float v2f __attribute__((ext_vector_type(2)));
typedef float v8f __attribute__((ext_vector_type(8)));

#define FDIM   256   // H*C for both layers
#define NHEAD  8
#define HCH    32

// ---- ordered-float <-> uint mapping for atomicMax-based segment max ----
__device__ __forceinline__ unsigned fflip(float f) {
    unsigned u = __float_as_uint(f);
    return (u & 0x80000000u) ? ~u : (u | 0x80000000u);
}
__device__ __forceinline__ float funflip(unsigned u) {
    return __uint_as_float((u & 0x80000000u) ? (u ^ 0x80000000u) : ~u);
}

// ---- init: zero accumulator + per-(node,head) max/sum state ----
__global__ void k_prep(float* __restrict__ acc, unsigned* __restrict__ mx,
                       float* __restrict__ sx, long long nAcc, long long nMh) {
    long long t = (long long)blockIdx.x * blockDim.x + threadIdx.x;
    if (t < nAcc) acc[t] = 0.0f;
    if (t < nMh) { mx[t] = 0u; sx[t] = 0.0f; }   // flipped 0u == below any finite float
}

// ---- WMMA f32 GEMM: H[n,256] = X[n,K] @ W[K,256] -----------------------
// 512 threads = 16 waves; wave w computes the 16x16 tile at columns [16w,16w+16)
// A tile (16 rows x K) staged in LDS once per block, shared by all 16 waves.
__global__ __launch_bounds__(512) void k_gemm_wmma(
        const float* __restrict__ X, const float* __restrict__ W,
        float* __restrict__ H, int nrows, int K) {
    extern __shared__ float lds[];                 // 16*K floats (<=16KB)
    const int m0  = blockIdx.x * 16;
    const int tid = threadIdx.x;

    for (int idx = tid; idx < 16 * K; idx += 512) {
        int r = idx / K, c = idx - r * K;
        int gm = m0 + r;
        lds[idx] = (gm < nrows) ? X[(long long)gm * K + c] : 0.0f;
    }
    __syncthreads();

    const int lane  = tid & 31;
    const int wave  = tid >> 5;
    const int n0    = wave * 16;
    const int row   = lane & 15;       // M row (A) / N col (B,C)
    const int khalf = lane >> 4;       // 0: K={0,1}, 1: K={2,3} within a step

    v8f acc = {};
    for (int k = 0; k < K; k += 4) {
        v2f a, b;
        // A 16x4 f32 layout: lanes 0-15 hold K=k,k+1; lanes 16-31 hold K=k+2,k+3
        a.x = lds[row * K + k + khalf * 2];
        a.y = lds[row * K + k + khalf * 2 + 1];
        // B 4x16 f32 layout (mirror of A): VGPR0->K even, VGPR1->K odd per half
        const float* wp = W + (long long)(k + khalf * 2) * FDIM + n0 + row;
        b.x = wp[0];
        b.y = wp[FDIM];
        acc = __builtin_amdgcn_wmma_f32_16x16x4_f32(
                  false, a, false, b, (short)0, acc, false, false);
    }

    // C/D layout: VGPR i -> M = i + 8*(lane>=16), N = lane&15
    const int mb = m0 + khalf * 8;
#pragma unroll
    for (int i = 0; i < 8; ++i) {
        int m = mb + i;
        if (m < nrows) H[(long long)m * FDIM + n0 + row] = acc[i];
    }
}

// ---- per-(node,head) attention dot products ---------------------------
__global__ void k_dots(const float* __restrict__ H,
                       const float* __restrict__ a_src, const float* __restrict__ a_dst,
                       float* __restrict__ ALs, float* __restrict__ ALd, int n) {
    long long t = (long long)blockIdx.x * blockDim.x + threadIdx.x;
    if (t >= (long long)n * NHEAD) return;
    int node = (int)(t >> 3), h = (int)(t & 7);
    const float4* hp = (const float4*)(H + (long long)node * FDIM + h * HCH);
    const float4* ap = (const float4*)(a_src + h * HCH);
    const float4* bp = (const float4*)(a_dst + h * HCH);
    float s = 0.f, d = 0.f;
#pragma unroll
    for (int i = 0; i < 8; ++i) {
        float4 hv = hp[i], av = ap[i], bv = bp[i];
        s += hv.x*av.x + hv.y*av.y + hv.z*av.z + hv.w*av.w;
        d += hv.x*bv.x + hv.y*bv.y + hv.z*bv.z + hv.w*bv.w;
    }
    ALs[t] = s; ALd[t] = d;
}

__device__ __forceinline__ float leaky02(float v) { return v > 0.f ? v : 0.2f * v; }

// ---- segment max over incoming edges ----------------------------------
__global__ void k_edge_max(const long long* __restrict__ src, const long long* __restrict__ dst,
                           const float* __restrict__ ALs, const float* __restrict__ ALd,
                           unsigned* __restrict__ Mx, long long etot) {
    long long t = (long long)blockIdx.x * blockDim.x + threadIdx.x;
    if (t >= etot * NHEAD) return;
    long long e = t >> 3; int h = (int)(t & 7);
    long long s = src[e], d = dst[e];
    float v = leaky02(ALs[s * NHEAD + h] + ALd[d * NHEAD + h]);
    atomicMax(&Mx[d * NHEAD + h], fflip(v));
}

// ---- exp(e - m) and segment sum ---------------------------------------
__global__ void k_edge_exp(const long long* __restrict__ src, const long long* __restrict__ dst,
                           const float* __restrict__ ALs, const float* __restrict__ ALd,
                           const unsigned* __restrict__ Mx, float* __restrict__ Sx,
                           float* __restrict__ Eb, long long etot) {
    long long t = (long long)blockIdx.x * blockDim.x + threadIdx.x;
    if (t >= etot * NHEAD) return;
    long long e = t >> 3; int h = (int)(t & 7);
    long long s = src[e], d = dst[e];
    float v  = leaky02(ALs[s * NHEAD + h] + ALd[d * NHEAD + h]);
    float m  = funflip(Mx[d * NHEAD + h]);
    float ee = __expf(v - m);
    Eb[t] = ee;
    atomicAdd(&Sx[d * NHEAD + h], ee);
}

// ---- weighted scatter: ACC[dst] += alpha * H[src] ---------------------
// one block (256 threads) per edge; alpha per head in shared memory
__global__ __launch_bounds__(256) void k_edge_scatter(
        const long long* __restrict__ src, const long long* __restrict__ dst,
        const float* __restrict__ Eb, const float* __restrict__ Sx,
        const float* __restrict__ H, float* __restrict__ ACC, long long etot) {
    __shared__ float salpha[NHEAD];
    __shared__ long long ss, sd;
    long long e = blockIdx.x;
    int f = threadIdx.x;
    if (f == 0) { ss = src[e]; sd = dst[e]; }
    __syncthreads();
    if (f < NHEAD) salpha[f] = Eb[e * NHEAD + f] / (Sx[sd * NHEAD + f] + 1e-16f);
    __syncthreads();
    atomicAdd(&ACC[sd * FDIM + f], salpha[f >> 5] * H[ss * FDIM + f]);
}

// ---- bias + ELU in place ----------------------------------------------
__global__ void k_finish(float* __restrict__ buf, const float* __restrict__ bias,
                         long long count) {
    long long t = (long long)blockIdx.x * blockDim.x + threadIdx.x;
    if (t >= count) return;
    float v = buf[t] + bias[t & (FDIM - 1)];
    buf[t] = v > 0.f ? v : (__expf(v) - 1.0f);
}

extern "C" void kernel_launch(void* const* d_in, const int* in_sizes, int n_in,
                              void* d_out, int out_size, void* d_ws, size_t ws_size,
                              hipStream_t stream) {
    (void)n_in; (void)out_size; (void)ws_size;
    const float*     x    = (const float*)d_in[0];
    const long long* ei   = (const long long*)d_in[1];   // int64 edge_index [2, Etot]
    const float*     W1   = (const float*)d_in[2];
    const float*     a1s  = (const float*)d_in[3];
    const float*     a1d  = (const float*)d_in[4];
    const float*     b1   = (const float*)d_in[5];
    const float*     W2   = (const float*)d_in[6];
    const float*     a2s  = (const float*)d_in[7];
    const float*     a2d  = (const float*)d_in[8];
    const float*     b2   = (const float*)d_in[9];

    const int       n    = in_sizes[0] / 128;          // IN = 128
    const long long etot = (long long)in_sizes[1] / 2;
    const long long* src = ei;
    const long long* dsti = ei + etot;

    // workspace carving (256B aligned)
    char*  ws  = (char*)d_ws;
    size_t off = 0;
    auto carve = [&](size_t bytes) -> void* {
        void* p = ws + off; off += (bytes + 255) & ~(size_t)255; return p;
    };
    float*    H   = (float*)   carve((size_t)n * FDIM * sizeof(float));
    float*    ACC = (float*)   carve((size_t)n * FDIM * sizeof(float));
    float*    ALs = (float*)   carve((size_t)n * NHEAD * sizeof(float));
    float*    ALd = (float*)   carve((size_t)n * NHEAD * sizeof(float));
    unsigned* Mx  = (unsigned*)carve((size_t)n * NHEAD * sizeof(unsigned));
    float*    Sx  = (float*)   carve((size_t)n * NHEAD * sizeof(float));
    float*    Eb  = (float*)   carve((size_t)etot * NHEAD * sizeof(float));

    const long long nAcc = (long long)n * FDIM;
    const long long nMh  = (long long)n * NHEAD;
    const long long eth  = etot * NHEAD;
    const int TB = 256;
    const unsigned gAcc  = (unsigned)((nAcc + TB - 1) / TB);
    const unsigned gMh   = (unsigned)((nMh  + TB - 1) / TB);
    const unsigned gEth  = (unsigned)((eth  + TB - 1) / TB);
    const unsigned gGemm = (unsigned)((n + 15) / 16);
    float* out = (float*)d_out;

    // ---------------- layer 1 ----------------
    k_prep<<<gAcc, TB, 0, stream>>>(ACC, Mx, Sx, nAcc, nMh);
    k_gemm_wmma<<<gGemm, 512, 16 * 128 * sizeof(float), stream>>>(x, W1, H, n, 128);
    k_dots<<<gMh, TB, 0, stream>>>(H, a1s, a1d, ALs, ALd, n);
    k_edge_max<<<gEth, TB, 0, stream>>>(src, dsti, ALs, ALd, Mx, etot);
    k_edge_exp<<<gEth, TB, 0, stream>>>(src, dsti, ALs, ALd, Mx, Sx, Eb, etot);
    k_edge_scatter<<<(unsigned)etot, 256, 0, stream>>>(src, dsti, Eb, Sx, H, ACC, etot);
    k_finish<<<gAcc, TB, 0, stream>>>(ACC, b1, nAcc);   // ACC = elu(out1 + b1) = layer-2 input

    // ---------------- layer 2 ----------------
    k_prep<<<gAcc, TB, 0, stream>>>(out, Mx, Sx, nAcc, nMh);
    k_gemm_wmma<<<gGemm, 512, 16 * 256 * sizeof(float), stream>>>(ACC, W2, H, n, 256);
    k_dots<<<gMh, TB, 0, stream>>>(H, a2s, a2d, ALs, ALd, n);
    k_edge_max<<<gEth, TB, 0, stream>>>(src, dsti, ALs, ALd, Mx, etot);
    k_edge_exp<<<gEth, TB, 0, stream>>>(src, dsti, ALs, ALd, Mx, Sx, Eb, etot);
    k_edge_scatter<<<(unsigned)etot, 256, 0, stream>>>(src, dsti, Eb, Sx, H, out, etot);
    k_finish<<<gAcc, TB, 0, stream>>>(out, b2, nAcc);
}